// AdaptiveDiscriminatorAugmentation_46110768890070
// MI455X (gfx1250) — compile-verified
//
#include <hip/hip_runtime.h>
#include <math.h>

// Problem constants from the reference setup: [B=32, C=3, T=8, H=256, W=256].
namespace {
constexpr int B_ = 32, CT_ = 24, H_ = 256, W_ = 256;
constexpr int PLANE = H_ * W_;
constexpr long IMG = (long)CT_ * PLANE;
constexpr float CX = 0.5f * W_, CY = 0.5f * H_;
constexpr float DEG2RAD = 3.14159265358979323846f / 180.0f;

__device__ __forceinline__ float reflectf(float c, int size) {
  // _reflect: m = size-1; c = |c| mod 2m; min(c, 2m-c)
  const float m2 = 2.0f * (float)(size - 1);
  c = fabsf(c);
  c = fmodf(c, m2);
  return fminf(c, m2 - c);
}

// Composed index permutation sigma: output coord of (flip -> rot90 -> roll)
// chain mapped back to the ORIGINAL image coordinate.
__device__ __forceinline__ void sigma_map(int& y, int& x, bool ftrans, bool frot,
                                          bool fflip, int k, int sh, int sw) {
  if (ftrans) {                       // roll(shift): out[y,x] = in[(y-sh)%H,(x-sw)%W]
    y = ((y - sh) % H_ + H_) % H_;
    x = ((x - sw) % W_ + W_) % W_;
  }
  if (frot) {                         // jnp.rot90(k): k=1 -> in[x, W-1-y], etc.
    const int kk = ((k % 4) + 4) % 4;
    int ny = y, nx = x;
    if (kk == 1)      { ny = x;          nx = W_ - 1 - y; }
    else if (kk == 2) { ny = H_ - 1 - y; nx = W_ - 1 - x; }
    else if (kk == 3) { ny = H_ - 1 - x; nx = y;          }
    y = ny; x = nx;
  }
  if (fflip) x = W_ - 1 - x;          // x[..., ::-1]
}
} // namespace

// ---------------- CDNA5 async global->LDS path (guarded) ----------------
#if defined(__HIP_DEVICE_COMPILE__) && defined(__gfx1250__) && \
    __has_builtin(__builtin_amdgcn_global_load_async_to_lds_b32)
#define ADA_ASYNC_LDS 1
#else
#define ADA_ASYNC_LDS 0
#endif

#if ADA_ASYNC_LDS
typedef __attribute__((address_space(1))) int g1i_t;  // global (AS1) int
typedef __attribute__((address_space(3))) int l3i_t;  // LDS (AS3) int

__device__ __forceinline__ void async_g2l_b32(const float* g, float* l) {
  // Cast through integers: flat LDS address truncates to the AS3 byte offset
  // (low 32 bits of the flat address); global flat == AS1 address.
  g1i_t* gp = (g1i_t*)(unsigned long long)(uintptr_t)g;
  l3i_t* lp = (l3i_t*)(unsigned int)(uintptr_t)l;
  __builtin_amdgcn_global_load_async_to_lds_b32(gp, lp, 0, 0);
}

template <int N>
__device__ __forceinline__ void wait_async() {
#if __has_builtin(__builtin_amdgcn_s_wait_asynccnt)
  __builtin_amdgcn_s_wait_asynccnt(N);
#else
  asm volatile("s_wait_asynccnt %0" ::"i"(N) : "memory");
#endif
}
#endif

// Pass 0: sigma (flip/rot90/roll) fused with the scale1 warp (or exact
// permutation copy when mask_scale1 is false). Destination buffer chosen by
// parity of remaining active warps so the chain always terminates in d_out.
__global__ void __launch_bounds__(256) ada_pass0(
    const float* __restrict__ in, float* __restrict__ outbuf, float* __restrict__ wsbuf,
    const int* __restrict__ mflip, const int* __restrict__ mrot90,
    const int* __restrict__ mtrans, const int* __restrict__ mscale1,
    const int* __restrict__ mrot1, const int* __restrict__ mscale2,
    const int* __restrict__ mrot2, const float* __restrict__ scale1,
    const int* __restrict__ rot90_k, const int* __restrict__ shift_h,
    const int* __restrict__ shift_w)
{
  const int b   = blockIdx.z;
  const int tid = threadIdx.x;
  const int x   = blockIdx.x * 32 + (tid & 31);
  const int y   = blockIdx.y * 8  + (tid >> 5);

  const bool ff = mflip[b]   != 0;
  const bool fr = mrot90[b]  != 0;
  const bool ft = mtrans[b]  != 0;
  const bool a0 = mscale1[b] != 0;
  const int  r1 = (mrot1[b] != 0) + (mscale2[b] != 0) + (mrot2[b] != 0);
  float* __restrict__ dst = ((r1 & 1) == 0) ? outbuf : wsbuf;

  const int k = rot90_k[0], sh = shift_h[0], sw = shift_w[0];
  const long ibase = (long)b * IMG;
  const int  opix  = y * W_ + x;

  if (a0) {
    const float inv = 1.0f / scale1[b];   // isotropic, angle = 0
    float xi = ((float)x - CX) * inv + CX;
    float yi = ((float)y - CY) * inv + CY;
    xi = reflectf(xi, W_); yi = reflectf(yi, H_);
    int x0 = (int)floorf(xi), y0 = (int)floorf(yi);
    const float wx = xi - (float)x0, wy = yi - (float)y0;  // weights before clip (matches ref)
    x0 = min(max(x0, 0), W_ - 1);
    y0 = min(max(y0, 0), H_ - 1);
    const int x1 = min(x0 + 1, W_ - 1);
    const int y1 = min(y0 + 1, H_ - 1);
    int ya = y0, xa = x0; sigma_map(ya, xa, ft, fr, ff, k, sh, sw);
    int yb = y0, xb = x1; sigma_map(yb, xb, ft, fr, ff, k, sh, sw);
    int yc = y1, xc = x0; sigma_map(yc, xc, ft, fr, ff, k, sh, sw);
    int yd = y1, xd = x1; sigma_map(yd, xd, ft, fr, ff, k, sh, sw);
    const int o00 = ya * W_ + xa, o01 = yb * W_ + xb;
    const int o10 = yc * W_ + xc, o11 = yd * W_ + xd;
    const float w00 = (1.f - wy) * (1.f - wx), w01 = (1.f - wy) * wx;
    const float w10 = wy * (1.f - wx),         w11 = wy * wx;
#pragma unroll 4
    for (int c = 0; c < CT_; ++c) {
      const float* __restrict__ p = in + ibase + (long)c * PLANE;
      if (c + 1 < CT_) {  // gfx1250 global_prefetch_b8: hide HBM latency, next channel
        __builtin_prefetch(p + PLANE + o00, 0, 0);
        __builtin_prefetch(p + PLANE + o10, 0, 0);
      }
      dst[ibase + (long)c * PLANE + opix] =
          p[o00] * w00 + p[o01] * w01 + p[o10] * w10 + p[o11] * w11;
    }
  } else {
    int ys = y, xs = x; sigma_map(ys, xs, ft, fr, ff, k, sh, sw);
    const int os = ys * W_ + xs;
#pragma unroll 4
    for (int c = 0; c < CT_; ++c)
      dst[ibase + (long)c * PLANE + opix] = in[ibase + (long)c * PLANE + os];
  }
}

// Passes 1..3: rot1 / scale2 / rot2. Workgroups of inactive batch elements
// exit immediately (data stays in place; buffer parity accounts for it).
// Channel loop is a double-buffered async-to-LDS software pipeline: channel
// c+1's four gathers ride the ASYNCcnt path while channel c is blended.
__global__ void __launch_bounds__(256) ada_warp(
    float* __restrict__ outbuf, float* __restrict__ wsbuf,
    const int* __restrict__ mrot1, const int* __restrict__ mscale2,
    const int* __restrict__ mrot2, const float* __restrict__ angle1,
    const float* __restrict__ scale2, const float* __restrict__ angle2,
    int pass_id)
{
  const int b = blockIdx.z;
  const bool a1 = mrot1[b] != 0, a2 = mscale2[b] != 0, a3 = mrot2[b] != 0;
  const bool act = (pass_id == 1) ? a1 : (pass_id == 2) ? a2 : a3;
  if (!act) return;

  // #active warps in passes >= pass_id; data sits in d_out iff this is even.
  const int rk = (pass_id == 1) ? (int)a1 + (int)a2 + (int)a3
               : (pass_id == 2) ? (int)a2 + (int)a3
                                : (int)a3;
  const float* __restrict__ src = ((rk & 1) == 0) ? outbuf : wsbuf;
  float*       __restrict__ dst = ((rk & 1) == 0) ? wsbuf  : outbuf;

  float cth = 1.f, sth = 0.f, invx = 1.f, invy = 1.f;
  if (pass_id == 1)      { const float t = angle1[b] * DEG2RAD; cth = cosf(t); sth = sinf(t); }
  else if (pass_id == 2) { invy = 1.0f / scale2[2 * b + 0]; invx = 1.0f / scale2[2 * b + 1]; }
  else                   { const float t = angle2[b] * DEG2RAD; cth = cosf(t); sth = sinf(t); }

  const int tid = threadIdx.x;
  const int x = blockIdx.x * 32 + (tid & 31);
  const int y = blockIdx.y * 8  + (tid >> 5);
  const float dx = (float)x - CX, dy = (float)y - CY;
  float xi = ( cth * dx + sth * dy) * invx + CX;
  float yi = (-sth * dx + cth * dy) * invy + CY;
  xi = reflectf(xi, W_); yi = reflectf(yi, H_);
  int x0 = (int)floorf(xi), y0 = (int)floorf(yi);
  const float wx = xi - (float)x0, wy = yi - (float)y0;
  x0 = min(max(x0, 0), W_ - 1);
  y0 = min(max(y0, 0), H_ - 1);
  const int x1 = min(x0 + 1, W_ - 1);
  const int y1 = min(y0 + 1, H_ - 1);
  const int o00 = y0 * W_ + x0, o01 = y0 * W_ + x1;
  const int o10 = y1 * W_ + x0, o11 = y1 * W_ + x1;
  const float w00 = (1.f - wy) * (1.f - wx), w01 = (1.f - wy) * wx;
  const float w10 = wy * (1.f - wx),         w11 = wy * wx;
  const long ibase = (long)b * IMG;
  const int  opix  = y * W_ + x;

#if ADA_ASYNC_LDS
  __shared__ float stage[2][4 * 256];          // 8 KB double buffer
  float* slot0 = &stage[0][tid * 4];
  float* slot1 = &stage[1][tid * 4];
  {
    const float* p0 = src + ibase;             // prologue: channel 0 in flight
    async_g2l_b32(p0 + o00, slot0 + 0);
    async_g2l_b32(p0 + o01, slot0 + 1);
    async_g2l_b32(p0 + o10, slot0 + 2);
    async_g2l_b32(p0 + o11, slot0 + 3);
  }
  for (int c = 0; c < CT_; ++c) {
    float* cur = (c & 1) ? slot1 : slot0;
    float* nxt = (c & 1) ? slot0 : slot1;
    if (c + 1 < CT_) {
      const float* pn = src + ibase + (long)(c + 1) * PLANE;
      async_g2l_b32(pn + o00, nxt + 0);
      async_g2l_b32(pn + o01, nxt + 1);
      async_g2l_b32(pn + o10, nxt + 2);
      async_g2l_b32(pn + o11, nxt + 3);
      wait_async<4>();   // in-order completion: channel c's 4 loads are done
    } else {
      wait_async<0>();
    }
    dst[ibase + (long)c * PLANE + opix] =
        cur[0] * w00 + cur[1] * w01 + cur[2] * w10 + cur[3] * w11;
  }
#else
#pragma unroll 4
  for (int c = 0; c < CT_; ++c) {
    const float* __restrict__ p = src + ibase + (long)c * PLANE;
    if (c + 1 < CT_) {
      __builtin_prefetch(p + PLANE + o00, 0, 0);
      __builtin_prefetch(p + PLANE + o10, 0, 0);
    }
    dst[ibase + (long)c * PLANE + opix] =
        p[o00] * w00 + p[o01] * w01 + p[o10] * w10 + p[o11] * w11;
  }
#endif
}

extern "C" void kernel_launch(void* const* d_in, const int* in_sizes, int n_in,
                              void* d_out, int out_size, void* d_ws, size_t ws_size,
                              hipStream_t stream) {
  (void)in_sizes; (void)n_in; (void)out_size; (void)ws_size;
  const float* images  = (const float*)d_in[0];
  const int*   mflip   = (const int*)d_in[1];
  const int*   mrot90  = (const int*)d_in[2];
  const int*   mtrans  = (const int*)d_in[3];
  const int*   mscale1 = (const int*)d_in[4];
  const int*   mrot1   = (const int*)d_in[5];
  const int*   mscale2 = (const int*)d_in[6];
  const int*   mrot2   = (const int*)d_in[7];
  const float* scale1  = (const float*)d_in[8];
  const float* angle1  = (const float*)d_in[9];
  const float* scale2  = (const float*)d_in[10];
  const float* angle2  = (const float*)d_in[11];
  const int*   rot90_k = (const int*)d_in[12];
  const int*   shift_h = (const int*)d_in[13];
  const int*   shift_w = (const int*)d_in[14];

  float* out = (float*)d_out;
  float* ws  = (float*)d_ws;   // needs one image tensor (192 MiB) of scratch

  dim3 block(256);                         // 8 wave32 waves per workgroup
  dim3 grid(W_ / 32, H_ / 8, B_);          // 32x8 pixel tile per block

  ada_pass0<<<grid, block, 0, stream>>>(images, out, ws, mflip, mrot90, mtrans,
                                        mscale1, mrot1, mscale2, mrot2, scale1,
                                        rot90_k, shift_h, shift_w);
  for (int p = 1; p <= 3; ++p)
    ada_warp<<<grid, block, 0, stream>>>(out, ws, mrot1, mscale2, mrot2,
                                         angle1, scale2, angle2, p);
}